// PNNLayer_70592082477577
// MI455X (gfx1250) — compile-verified
//
#include <hip/hip_runtime.h>

typedef __attribute__((ext_vector_type(2))) float v2f;
typedef __attribute__((ext_vector_type(8))) float v8f;

#define DD    64          // D
#define AA    32          // A
#define KTOT  96          // A + D
#define KP    48          // KTOT/2 interleaved K-pair rows
#define BP2   80          // LDS pair-row stride in float2 (160 dwords = 32 mod 64 banks)

#define WMMA_F32(acc, av, bv)                                                   \
  acc = __builtin_amdgcn_wmma_f32_16x16x4_f32(false, (av), false, (bv),         \
                                              (short)0, (acc), false, false)

// ---------------------------------------------------------------------------
// Kernel 1: build the fused B matrix (96 x 64) in workspace (plain layout).
//   rows 0..31 : B[a][k] = (1/A) * sum_d embeds[aid[a]][d] * W_hidden[k][d]
//   rows 32..95: B[32+d][k] = W_hidden[k][64+d]   (= W2^T)
// ---------------------------------------------------------------------------
__global__ __launch_bounds__(256) void pnn_prep(
    const float* __restrict__ embeds, const int* __restrict__ anchor,
    const float* __restrict__ Wh, float* __restrict__ Bws) {
  int t = blockIdx.x * blockDim.x + threadIdx.x;
  if (t < AA * DD) {
    int a = t >> 6, k = t & 63;
    const float* e = embeds + (size_t)anchor[a] * DD;
    const float* w = Wh + (size_t)k * (2 * DD);
    float acc = 0.f;
#pragma unroll
    for (int d = 0; d < DD; ++d) acc += e[d] * w[d];
    Bws[a * DD + k] = acc * (1.0f / AA);
  } else if (t < AA * DD + DD * DD) {
    int i = t - AA * DD;
    int d = i >> 6, k = i & 63;
    Bws[(AA + d) * DD + k] = Wh[(size_t)k * (2 * DD) + DD + d];
  }
}

// ---------------------------------------------------------------------------
// Kernel 2: out(N x 64) = [dists^T | embeds](N x 96) @ B(96 x 64) + b_hidden
// One wave per 16-row output tile; 4 waves per block.
// B cached in LDS, K-pair interleaved: Bs[kp][col] = {B[2kp][col], B[2kp+1][col]}
// so each WMMA B-fragment is one aligned ds_load_b64 into an even VGPR pair.
// ---------------------------------------------------------------------------
__global__ __launch_bounds__(128) void pnn_main(
    const float* __restrict__ embeds,   // (N, 64) row-major
    const float* __restrict__ dists,    // (32, N) row-major
    const float* __restrict__ Bws,      // (96, 64) in workspace
    const float* __restrict__ bh,       // (64,)
    float* __restrict__ out,            // (N, 64)
    int Nn, int ntiles) {
  __shared__ v2f Bs[KP * BP2];          // 30 KB
  {
    float* Bsf = (float*)Bs;
    for (int i = threadIdx.x; i < KTOT * DD; i += blockDim.x) {
      int row = i >> 6, col = i & 63;
      Bsf[((row >> 1) * BP2 + col) * 2 + (row & 1)] = Bws[i];
    }
  }
  __syncthreads();

  const int wave = threadIdx.x >> 5;
  const int lane = threadIdx.x & 31;
  const int tile = blockIdx.x * 4 + wave;
  if (tile >= ntiles) return;          // wave-uniform: EXEC stays all-ones

  const int n0   = tile << 4;          // first output row of this tile
  const int half = lane >> 4;          // 0: K={0,1}, 1: K={2,3} within chunk
  const int lrow = lane & 15;          // A: M index / B,C,D: N index

  // C/D 16x16 f32 layout: vgpr r, lanes0-15 -> M=r, lanes16-31 -> M=r+8; N=lane&15.
  // b_hidden depends only on N -> broadcast into each accumulator column group.
  v8f c0, c1, c2, c3;
  {
    const float b0 = bh[lrow], b1 = bh[16 + lrow], b2 = bh[32 + lrow], b3 = bh[48 + lrow];
#pragma unroll
    for (int r = 0; r < 8; ++r) { c0[r] = b0; c1[r] = b1; c2[r] = b2; c3[r] = b3; }
  }

#pragma unroll
  for (int kb = 0; kb < KTOT; kb += 4) {
    const int k0 = kb + 2 * half;      // this lane's K pair {k0, k0+1}

    // A-fragment (16x4 f32): lane holds A[M=lrow][k0], A[M=lrow][k0+1]
    v2f a;
    if (kb < AA) {
      // distsT[n][k] = dists[k*N + n] -- coalesced in n across the half-wave
      a.x = dists[(size_t)(k0 + 0) * Nn + n0 + lrow];
      a.y = dists[(size_t)(k0 + 1) * Nn + n0 + lrow];
    } else {
      // embeds row (n0+lrow), cols (k0-32, k0-31): one 8B load; the row's
      // 256B is fully consumed across the 16 embed k-chunks (cache reuse)
      a = *(const v2f*)(embeds + (size_t)(n0 + lrow) * DD + (k0 - AA));
    }

    // B-fragments: single b64 load per n-tile; lanes 0-15 read pair-row kp,
    // lanes 16-31 read kp+1 (disjoint bank halves with BP2=80 stride).
    const int kp = (kb >> 1) + half;
    const v2f* bp = Bs + kp * BP2 + lrow;
    const v2f b0 = bp[0];
    const v2f b1 = bp[16];
    const v2f b2 = bp[32];
    const v2f b3 = bp[48];
    WMMA_F32(c0, a, b0);
    WMMA_F32(c1, a, b1);
    WMMA_F32(c2, a, b2);
    WMMA_F32(c3, a, b3);
  }

  // Store: vgpr r -> output row n0 + r + 8*half, column = 16*ntile + lrow.
  // Each half-wave writes 64B contiguous per row.
#pragma unroll
  for (int r = 0; r < 8; ++r) {
    float* o = out + (size_t)(n0 + r + 8 * half) * DD + lrow;
    o[0]  = c0[r];
    o[16] = c1[r];
    o[32] = c2[r];
    o[48] = c3[r];
  }
}

// ---------------------------------------------------------------------------
extern "C" void kernel_launch(void* const* d_in, const int* in_sizes, int n_in,
                              void* d_out, int out_size, void* d_ws, size_t ws_size,
                              hipStream_t stream) {
  const float* embeds = (const float*)d_in[0];   // (N, 64) f32
  const float* dists  = (const float*)d_in[1];   // (32, N) f32
  const int*   aid    = (const int*)  d_in[2];   // (32,)   i32
  const float* Wh     = (const float*)d_in[3];   // (64,128) f32
  const float* bh     = (const float*)d_in[4];   // (64,)   f32
  float*       out    = (float*)d_out;           // (N, 64) f32
  float*       Bws    = (float*)d_ws;            // 96*64 f32 = 24 KB scratch

  const int Nn     = in_sizes[0] / DD;           // 50000
  const int ntiles = Nn / 16;                    // 3125 (exact)

  const int prep_threads = AA * DD + DD * DD;    // 6144
  pnn_prep<<<(prep_threads + 255) / 256, 256, 0, stream>>>(embeds, aid, Wh, Bws);

  pnn_main<<<(ntiles + 3) / 4, 128, 0, stream>>>(embeds, dists, Bws, bh, out,
                                                 Nn, ntiles);
}